// HDLoss_31602369364236
// MI455X (gfx1250) — compile-verified
//
#include <hip/hip_runtime.h>

#define HW 512
#define NPIX (HW * HW)
#define BIGF 1.0e6f
#define CLIPF 20.0f
#define EPSF 1.0e-4f
#define NEGBIG (-3.0e38f)

typedef __attribute__((ext_vector_type(2))) float v2f;
typedef __attribute__((ext_vector_type(8))) float v8f;

// ---- wave-level f32 sum via V_WMMA_F32_16X16X4_F32 with all-ones A --------
// D[m,n] = sum_k A[m,k]*B[k,n] + C[m,n]; with A==1, every column of D holds a
// partial column-sum of B; union over (k,n) covers all 64 B entries exactly
// once, so sum over the 16 columns of row 0 == sum of all B values.
// We place the wave's 32 per-lane values in B's first VGPR (second = 0).
__device__ __forceinline__ float wave_sum_wmma(float q) {
  v2f a; a[0] = 1.0f; a[1] = 1.0f;
  v2f b; b[0] = q;    b[1] = 0.0f;
  v8f c = {};
  c = __builtin_amdgcn_wmma_f32_16x16x4_f32(false, a, false, b, (short)0, c,
                                            false, false);
  int lane = (int)(threadIdx.x & 31u);
  // c[0]: lanes 0..15 hold D[0][n] (lanes 16..31 duplicate row M=8)
  float r = (lane < 16) ? c[0] : 0.0f;
  r += __shfl_xor(r, 8, 32);
  r += __shfl_xor(r, 4, 32);
  r += __shfl_xor(r, 2, 32);
  r += __shfl_xor(r, 1, 32);
  return r;  // lane 0 (indeed lanes 0..15) holds the full 32-lane sum
}

__device__ __forceinline__ float wave_max32(float q) {
  q = fmaxf(q, __shfl_xor(q, 16, 32));
  q = fmaxf(q, __shfl_xor(q, 8, 32));
  q = fmaxf(q, __shfl_xor(q, 4, 32));
  q = fmaxf(q, __shfl_xor(q, 2, 32));
  q = fmaxf(q, __shfl_xor(q, 1, 32));
  return q;
}

// ---- K1: axis-0 forward/backward min-plus scans -> g^2 for both masks -----
// dp-EDT mask = (target==0), dn-EDT mask = (target!=0). Coalesced row access.
__global__ void hd_colscan(const int* __restrict__ target,
                           float* __restrict__ g2p, float* __restrict__ g2n) {
  int j = blockIdx.x * blockDim.x + threadIdx.x;  // 0..511, exact
  float cp = BIGF, cn = BIGF;
  for (int i = 0; i < HW; ++i) {
    int idx = i * HW + j;
    bool fg = (target[idx] != 0);
    cp = (!fg) ? fminf(cp + 1.0f, BIGF) : 0.0f;
    cn = (fg)  ? fminf(cn + 1.0f, BIGF) : 0.0f;
    g2p[idx] = cp;   // stash forward scan values
    g2n[idx] = cn;
  }
  cp = BIGF; cn = BIGF;
  for (int i = HW - 1; i >= 0; --i) {
    int idx = i * HW + j;
    bool fg = (target[idx] != 0);
    cp = (!fg) ? fminf(cp + 1.0f, BIGF) : 0.0f;
    cn = (fg)  ? fminf(cn + 1.0f, BIGF) : 0.0f;
    float gp = fminf(g2p[idx], cp);
    float gn = fminf(g2n[idx], cn);
    g2p[idx] = gp * gp;  // squared axis-0 distance
    g2n[idx] = gn * gn;
  }
}

// ---- K2: row-wise envelope d(i,j) = sqrt(min_k g2[i,k] + (j-k)^2) ---------
// One block per row; g2 rows staged in LDS (broadcast reads). Writes the
// distance back IN PLACE over g2 (each block owns its row). Also emits a
// per-block partial max of values < CLIP for the clip step.
__global__ void hd_envelope(float* __restrict__ g2p, float* __restrict__ g2n,
                            float* __restrict__ pclip) {
  __shared__ float sp[HW];
  __shared__ float sn[HW];
  const int i = blockIdx.x;
  const int t = threadIdx.x;          // 0..255
  const int base = i * HW;
  sp[t] = g2p[base + t];       sp[t + 256] = g2p[base + t + 256];
  sn[t] = g2n[base + t];       sn[t + 256] = g2n[base + t + 256];
  __syncthreads();

  float b0p = 3.0e38f, b1p = 3.0e38f, b0n = 3.0e38f, b1n = 3.0e38f;
  const float j0 = (float)t, j1 = (float)(t + 256);
  float kk = 0.0f;
  for (int k = 0; k < HW; ++k) {
    float gp = sp[k], gn = sn[k];
    float d0 = j0 - kk, d1 = j1 - kk;
    float o0 = d0 * d0, o1 = d1 * d1;           // exact in f32 (|d|<512)
    b0p = fminf(b0p, gp + o0);
    b1p = fminf(b1p, gp + o1);
    b0n = fminf(b0n, gn + o0);
    b1n = fminf(b1n, gn + o1);
    kk += 1.0f;
  }
  float dp0 = sqrtf(b0p), dp1 = sqrtf(b1p);
  float dn0 = sqrtf(b0n), dn1 = sqrtf(b1n);
  g2p[base + t] = dp0;  g2p[base + t + 256] = dp1;  // now holds raw dp
  g2n[base + t] = dn0;  g2n[base + t + 256] = dn1;  // now holds raw dn

  // partial max of dp where dp < CLIP  (vmax = max(where(dp<20, dp, -inf)))
  float m = fmaxf((dp0 < CLIPF) ? dp0 : NEGBIG,
                  (dp1 < CLIPF) ? dp1 : NEGBIG);
  __syncthreads();                 // everyone done reading sp
  sp[t] = m;
  __syncthreads();
  for (int o = 128; o > 0; o >>= 1) {
    if (t < o) sp[t] = fmaxf(sp[t], sp[t + o]);
    __syncthreads();
  }
  if (t == 0) pclip[i] = sp[0];
}

// ---- K3: reduce 512 per-row clip-maxes -> vmax scalar ---------------------
__global__ void hd_vmax(const float* __restrict__ pclip,
                        float* __restrict__ vmaxp) {
  __shared__ float s[256];
  int t = threadIdx.x;
  s[t] = fmaxf(pclip[t], pclip[t + 256]);
  __syncthreads();
  for (int o = 128; o > 0; o >>= 1) {
    if (t < o) s[t] = fmaxf(s[t], s[t + o]);
    __syncthreads();
  }
  if (t == 0) vmaxp[0] = s[0];
}

// ---- K4: clip + log + masked partials (4 sums via WMMA, 2 maxes) ----------
__global__ void hd_elemwise(const float* __restrict__ inp,
                            const int* __restrict__ target,
                            const float* __restrict__ dpr,
                            const float* __restrict__ dnr,
                            const float* __restrict__ vmaxp,
                            float* __restrict__ ps1, float* __restrict__ ps2,
                            float* __restrict__ psbg, float* __restrict__ psfg,
                            float* __restrict__ pmp, float* __restrict__ pmn) {
  const float vmax = vmaxp[0];
  const int tid = blockIdx.x * blockDim.x + threadIdx.x;  // 0..65535
  float s1 = 0.f, s2 = 0.f, sbg = 0.f, sfg = 0.f, mp = 0.f, mn = 0.f;
  for (int p = tid; p < NPIX; p += 65536) {               // 4 uniform iters
    __builtin_prefetch(dpr + p + 65536, 0, 1);            // global_prefetch_b8
    float dp = dpr[p];
    dp = (dp > CLIPF) ? vmax : dp;
    float dpl = (dp != 0.0f) ? logf(dp) : 0.0f;           // log_nonzero
    float dn = dnr[p];
    float dnl = (dn != 0.0f) ? logf(dn) : 0.0f;
    float x = inp[p];
    int tg = target[p];
    s1 += x * dpl;                                  // sum(inp * dp_log)
    s2 += (1.0f - x) * dnl;                         // sum((1-inp) * dn_log)
    sbg += (tg == 0) ? x : 0.0f;                    // sum(inp | bg)
    sfg += (tg != 0) ? (1.0f - x) : 0.0f;           // sum(1-inp | fg)
    mp = fmaxf(mp, (x != 0.0f) ? dpl : 0.0f);       // d_max_p pre-EPS
    mn = fmaxf(mn, ((1.0f - x) != 0.0f) ? dnl : 0.0f);
  }
  // per-wave reductions (uniform control flow; EXEC all-ones for WMMA)
  float r1 = wave_sum_wmma(s1);
  float r2 = wave_sum_wmma(s2);
  float r3 = wave_sum_wmma(sbg);
  float r4 = wave_sum_wmma(sfg);
  float m1 = wave_max32(mp);
  float m2 = wave_max32(mn);

  __shared__ float red[48];
  int wid = threadIdx.x >> 5, lane = threadIdx.x & 31;
  if (lane == 0) {
    red[wid] = r1; red[8 + wid] = r2; red[16 + wid] = r3;
    red[24 + wid] = r4; red[32 + wid] = m1; red[40 + wid] = m2;
  }
  __syncthreads();
  if (threadIdx.x == 0) {
    float a = 0.f, b = 0.f, c = 0.f, d = 0.f, e = 0.f, f = 0.f;
    for (int w = 0; w < 8; ++w) {
      a += red[w]; b += red[8 + w]; c += red[16 + w]; d += red[24 + w];
      e = fmaxf(e, red[32 + w]); f = fmaxf(f, red[40 + w]);
    }
    ps1[blockIdx.x] = a;  ps2[blockIdx.x] = b;
    psbg[blockIdx.x] = c; psfg[blockIdx.x] = d;
    pmp[blockIdx.x] = e;  pmn[blockIdx.x] = f;
  }
}

// ---- K5: single-wave final reduce + scalar combine ------------------------
__global__ void hd_final(const float* __restrict__ ps1,
                         const float* __restrict__ ps2,
                         const float* __restrict__ psbg,
                         const float* __restrict__ psfg,
                         const float* __restrict__ pmp,
                         const float* __restrict__ pmn,
                         float* __restrict__ out) {
  int lane = threadIdx.x;  // blockDim == 32, one wave, EXEC all-ones
  float s1 = 0.f, s2 = 0.f, sbg = 0.f, sfg = 0.f, mp = 0.f, mn = 0.f;
  for (int i = lane; i < 256; i += 32) {
    s1 += ps1[i]; s2 += ps2[i]; sbg += psbg[i]; sfg += psfg[i];
    mp = fmaxf(mp, pmp[i]); mn = fmaxf(mn, pmn[i]);
  }
  s1 = wave_sum_wmma(s1);
  s2 = wave_sum_wmma(s2);
  sbg = wave_sum_wmma(sbg);
  sfg = wave_sum_wmma(sfg);
  mp = wave_max32(mp);
  mn = wave_max32(mn);
  if (lane == 0) {
    float loss1 = (s1 / (mp + EPSF)) / fmaxf(sbg, EPSF);
    float loss2 = (s2 / (mn + EPSF)) / fmaxf(sfg, EPSF);
    out[0] = loss1 + loss2;
  }
}

extern "C" void kernel_launch(void* const* d_in, const int* in_sizes, int n_in,
                              void* d_out, int out_size, void* d_ws,
                              size_t ws_size, hipStream_t stream) {
  const float* inp   = (const float*)d_in[0];
  const int*  target = (const int*)d_in[1];
  float* W = (float*)d_ws;
  // workspace layout (floats): ~2.11 MB total
  float* g2p   = W;                      // [NPIX]  g2 then raw dp (in-place)
  float* g2n   = W + NPIX;               // [NPIX]  g2 then raw dn (in-place)
  float* pclip = W + 2 * NPIX;           // [512]
  float* vmaxp = W + 2 * NPIX + 512;     // [1]
  float* ps1   = W + 2 * NPIX + 544;     // [256] x 6
  float* ps2   = ps1 + 256;
  float* psbg  = ps2 + 256;
  float* psfg  = psbg + 256;
  float* pmp   = psfg + 256;
  float* pmn   = pmp + 256;
  float* out   = (float*)d_out;

  hipLaunchKernelGGL(hd_colscan, dim3(2), dim3(256), 0, stream,
                     target, g2p, g2n);
  hipLaunchKernelGGL(hd_envelope, dim3(HW), dim3(256), 0, stream,
                     g2p, g2n, pclip);
  hipLaunchKernelGGL(hd_vmax, dim3(1), dim3(256), 0, stream, pclip, vmaxp);
  hipLaunchKernelGGL(hd_elemwise, dim3(256), dim3(256), 0, stream,
                     inp, target, g2p, g2n, vmaxp,
                     ps1, ps2, psbg, psfg, pmp, pmn);
  hipLaunchKernelGGL(hd_final, dim3(1), dim3(32), 0, stream,
                     ps1, ps2, psbg, psfg, pmp, pmn, out);
}